// HH_Gap_86157043958378
// MI455X (gfx1250) — compile-verified
//
#include <hip/hip_runtime.h>
#include <hip/hip_bf16.h>
#include <stdint.h>

// ---------------------------------------------------------------------------
// Hodgkin-Huxley scan on MI455X (gfx1250).
//   - 8192 independent trajectories (B=16 x L=512), one per thread.
//   - Serial over N=2048 steps; state in registers; native exp/rcp.
//   - Drive z[b,k,l] staged into LDS with the Tensor Data Mover (TDM),
//     double-buffered, synchronized with s_wait_tensorcnt + barriers.
//   - Last (partial) chunk peeled so the steady-state inner loop has a
//     constant trip count and no per-step bounds check on the serial chain.
// ---------------------------------------------------------------------------

#define HH_B  16
#define HH_N  2048
#define HH_L  512

#define LT 256          // neurons (columns) per workgroup
#define KT 32           // time rows per TDM chunk
#define NCHUNK (HH_N / KT)

typedef unsigned int v4u __attribute__((ext_vector_type(4)));
typedef int          v8i __attribute__((ext_vector_type(8)));
typedef int          v4i __attribute__((ext_vector_type(4)));

// HH constants
#define C_GNA  40.0f
#define C_GK   35.0f
#define C_GL   0.1f
#define C_ENA  55.0f
#define C_EK  -77.0f
#define C_EL  -65.0f
#define C_IAPP 1.0f
#define C_VT  -50.0f
#define C_KP   3.0f
#define C_DT   0.05f

__device__ __forceinline__ float fast_rcp(float x) {
    return __builtin_amdgcn_rcpf(x);
}

// den = 1 - exp(t); rate = (den==0) ? singular : num/den
__device__ __forceinline__ float safe_rate(float num, float t, float sing) {
    float den = 1.0f - __expf(t);
    return (den == 0.0f) ? sing : num * fast_rcp(den);
}

// Crank-Nicolson style gate update
__device__ __forceinline__ float gate_update(float a, float b, float g) {
    float s = 0.5f * C_DT * (a + b);
    return (a * C_DT + (1.0f - s) * g) * fast_rcp(1.0f + s);
}

// One HH step: advances (V, m, n, h) by DT under drive zk, returns T output.
__device__ __forceinline__ float hh_step(float zk, float& V, float& m,
                                         float& n, float& h) {
    // membrane update (semi-implicit)
    const float m3   = m * m * m;
    const float pow1 = C_GNA * m3 * h;
    const float n2   = n * n;
    const float pow2 = C_GK * n2 * n2;
    const float Gs   = 0.5f * C_DT * (pow1 + pow2 + C_GL);
    const float E    = pow1 * C_ENA + pow2 * C_EK + C_GL * C_EL;
    const float Vn   = (V * (1.0f - Gs) + C_DT * (E + C_IAPP + zk))
                       * fast_rcp(1.0f + Gs);

    // rate functions (6 independent exps -> in-thread ILP)
    const float aN = safe_rate( 0.02f  * (Vn - 25.0f), (25.0f - Vn) * (1.0f/9.0f), 0.18f);
    const float bN = safe_rate(-0.002f * (Vn - 25.0f), (Vn - 25.0f) * (1.0f/9.0f), 0.08f);
    const float aM = safe_rate( 0.182f * (Vn + 35.0f), (-Vn - 35.0f) * (1.0f/9.0f), 1.638f);
    const float bM = safe_rate(-0.124f * (Vn + 35.0f), (Vn + 35.0f) * (1.0f/9.0f), 1.16f);
    const float aH = 0.25f * __expf((-Vn - 90.0f) * (1.0f/12.0f));
    const float bH = 0.25f * __expf(( Vn + 34.0f) * (1.0f/12.0f));

    m = gate_update(aM, bM, m);
    n = gate_update(aN, bN, n);
    h = gate_update(aH, bH, h);
    V = Vn;

    // T = sigmoid((V - VT)/KP)
    return fast_rcp(1.0f + __expf((C_VT - V) * (1.0f / C_KP)));
}

// Issue one TDM 2D tile load: KT rows x LT floats, row stride HH_L floats,
// from global `gptr` into LDS byte offset `lds_off`.
__device__ __forceinline__ void tdm_issue(const float* gptr, uint32_t lds_off) {
#if defined(__has_builtin) && __has_builtin(__builtin_amdgcn_tensor_load_to_lds)
    uint64_t ga = (uint64_t)(uintptr_t)gptr;

    // D# group 0 (128b): count=1 | lds_addr | global_addr[56:0] | type=2
    v4u g0;
    g0.x = 1u;                                               // count=1, user mode
    g0.y = lds_off;                                          // LDS byte address
    g0.z = (uint32_t)ga;                                     // global_addr[31:0]
    g0.w = (uint32_t)((ga >> 32) & 0x1FFFFFFu) | (2u << 30); // [56:32] | type=2

    // D# group 1 (256b)
    uint32_t d0 = (2u << 16);                      // wg_mask=0, data_size=4B
    uint32_t d1 = ((uint32_t)HH_L << 16);          // tensor_dim0 low16 (=512)
    uint32_t d2 = ((uint32_t)(HH_B * HH_N) << 16); // dim0 hi16=0 | dim1 low16
    uint32_t d3 = ((uint32_t)LT << 16);            // dim1 hi16=0 | tile_dim0 = LT
    uint32_t d4 = (uint32_t)KT;                    // tile_dim1 = KT | tile_dim2 = 0
    uint32_t d5 = (uint32_t)HH_L;                  // tensor_dim0_stride low32
    uint32_t d6 = 0u;                              // stride0 hi16 | stride1 low16
    uint32_t d7 = 0u;                              // stride1 hi32
    v8i g1 = { (int)d0, (int)d1, (int)d2, (int)d3,
               (int)d4, (int)d5, (int)d6, (int)d7 };

    v4i gz4 = { 0, 0, 0, 0 };                      // groups 2/3 unused (2D)
    v8i gz8 = { 0, 0, 0, 0, 0, 0, 0, 0 };          // 5th descriptor group unused

    __builtin_amdgcn_tensor_load_to_lds(g0, g1, gz4, gz4, gz8, 0);
#else
    (void)gptr; (void)lds_off;
#endif
}

__device__ __forceinline__ void tdm_wait0() {
#if defined(__has_builtin) && __has_builtin(__builtin_amdgcn_s_wait_tensorcnt)
    __builtin_amdgcn_s_wait_tensorcnt(0);
#endif
}

__global__ __launch_bounds__(LT, 1)
void HH_Gap_86157043958378_kernel(const float* __restrict__ z,
                                  float* __restrict__ out) {
    __shared__ float zbuf[2][KT * LT];           // 2 x 32KB double buffer

    const int tid = threadIdx.x;
    const int b  = blockIdx.x >> 1;              // batch
    const int l0 = (blockIdx.x & 1) * LT;        // neuron-column base
    const int l  = l0 + tid;

    const float* zb = z + ((size_t)b * HH_N) * HH_L + l0;   // this block's z slice
    float* ob = out + ((size_t)b * HH_N) * HH_L + l;        // column in output

    const uint32_t lds0 = (uint32_t)(uintptr_t)&zbuf[0][0];
    const uint32_t lds1 = (uint32_t)(uintptr_t)&zbuf[1][0];

    // state registers
    float V = -70.0f, m = 0.0f, n = 0.0f, h = 1.0f;

    // step 0 output: sigmoid((V0 - VT)/KP)
    ob[0] = fast_rcp(1.0f + __expf((C_VT - V) * (1.0f / C_KP)));

    // prologue: start DMA of chunk 0 into buffer 0
    if (tid < 32) tdm_issue(zb, lds0);

    float* op = ob + HH_L;                       // output row pointer (k=1)

    // steady state: all chunks except the last run exactly KT steps
    for (int c = 0; c < NCHUNK - 1; ++c) {
        if (tid < 32) {
            tdm_wait0();                                     // chunk c landed
            tdm_issue(zb + (size_t)(c + 1) * KT * HH_L,      // prefetch chunk c+1
                      (c & 1) ? lds0 : lds1);
        }
        __syncthreads();                                     // publish chunk c

        const float* zc = &zbuf[c & 1][tid];
        #pragma unroll 4
        for (int kk = 0; kk < KT; ++kk) {
            const float T = hh_step(zc[kk * LT], V, m, n, h);
            op[0] = T;                                       // coalesced b32 store
            op += HH_L;
        }
        __syncthreads();    // all waves done reading buf[c&1] before reuse
    }

    // epilogue: last chunk drives steps (N-KT+1)..(N-1) -> KT-1 steps
    {
        const int c = NCHUNK - 1;
        if (tid < 32) tdm_wait0();
        __syncthreads();

        const float* zc = &zbuf[c & 1][tid];
        #pragma unroll 4
        for (int kk = 0; kk < KT - 1; ++kk) {
            const float T = hh_step(zc[kk * LT], V, m, n, h);
            op[0] = T;
            op += HH_L;
        }
    }
}

extern "C" void kernel_launch(void* const* d_in, const int* in_sizes, int n_in,
                              void* d_out, int out_size, void* d_ws, size_t ws_size,
                              hipStream_t stream) {
    (void)in_sizes; (void)n_in; (void)out_size; (void)d_ws; (void)ws_size;
    const float* z = (const float*)d_in[0];
    float* out = (float*)d_out;
    // 16 batches x (512/256) column tiles = 32 blocks of 256 threads (8 waves)
    HH_Gap_86157043958378_kernel<<<HH_B * (HH_L / LT), LT, 0, stream>>>(z, out);
}